// RoIAlign_60438779789713
// MI455X (gfx1250) — compile-verified
//
#include <hip/hip_runtime.h>

// ---------------------------------------------------------------------------
// Crop-and-resize (RoIAlign-like), fp32, MI455X / gfx1250.
//
// Memory-bound (~1 GB traffic -> ~44us floor @ 23.3 TB/s; ~0.3 GFLOP total,
// so matrix engines are irrelevant). Strategy:
//   * one thread per (box, crop position): tap offsets + bilinear weights
//     computed ONCE, reused across all channels (NCHW plane stride constant)
//   * 4 gathered loads per output, served mostly from L2 (box regions reused)
//   * non-temporal stores: keep the 205MB output stream out of L2 so the
//     275MB featuremap (with ~3x reuse) stays resident
//   * global_prefetch_b8 (gfx1250) for the tap lines 4 channel planes ahead
// ---------------------------------------------------------------------------

namespace {
constexpr int FM_N = 4;
constexpr int FM_C = 256;
constexpr int FM_H = 200;
constexpr int FM_W = 336;
constexpr int M_BOXES = 1024;
constexpr int CROP = 14;
constexpr int NPOS = CROP * CROP;          // 196
constexpr int CH_SPLIT = 8;                // channel slices per box
constexpr int C_PER_BLK = FM_C / CH_SPLIT; // 32
constexpr int HW = FM_H * FM_W;            // 67200 (channel plane stride)
constexpr int PF_DIST = 4;                 // prefetch distance in channels
}

__global__ __launch_bounds__(224) void crop_resize_gfx1250(
    const float* __restrict__ fm,     // (4,256,200,336)
    const float* __restrict__ boxes,  // (1024,4) = x1,y1,x2,y2
    const int*   __restrict__ box_ind,// (1024,)
    float*       __restrict__ out)    // (1024,256,14,14)
{
    const int m  = blockIdx.x;          // box
    const int cz = blockIdx.y;          // channel slice
    const int t  = threadIdx.x;         // crop position
    if (t >= NPOS) return;

    // --- box parameters (uniform across block -> scalar loads) -------------
    const float x1 = boxes[4 * m + 0];
    const float y1 = boxes[4 * m + 1];
    const float x2 = boxes[4 * m + 2];
    const float y2 = boxes[4 * m + 3];
    const int   b  = box_ind[m];

    const float Wm1 = (float)(FM_W - 1);  // 335
    const float Hm1 = (float)(FM_H - 1);  // 199

    // --- replicate reference math (incl. its spacing_w-in-nh quirk and the
    //     normalize -> denormalize rounding order) exactly -------------------
    const float sw  = (x2 - x1) / (float)CROP;
    const float sh  = (y2 - y1) / (float)CROP;
    const float nx0 = (x1 + sw * 0.5f - 0.5f) / Wm1;
    const float ny0 = (y1 + sh * 0.5f - 0.5f) / Hm1;
    const float nw  = sw * (float)(CROP - 1) / Wm1;
    const float nh  = sw * (float)(CROP - 1) / Hm1;   // reference uses sw here
    const float x2c = nx0 + nw;
    const float y2c = ny0 + nh;

    const int i = t / CROP;
    const int j = t - i * CROP;

    const float in_y = ny0 * Hm1 + (float)i * ((y2c - ny0) * Hm1 / (float)(CROP - 1));
    const float in_x = nx0 * Wm1 + (float)j * ((x2c - nx0) * Wm1 / (float)(CROP - 1));

    const bool valid = (in_y >= 0.0f) && (in_y <= Hm1) &&
                       (in_x >= 0.0f) && (in_x <= Wm1);

    const float y0f = floorf(in_y);
    const float x0f = floorf(in_x);
    const float yf  = in_y - y0f;
    const float xf  = in_x - x0f;

    const int y0i = (int)fminf(fmaxf(y0f,        0.0f), Hm1);
    const int y1i = (int)fminf(fmaxf(y0f + 1.0f, 0.0f), Hm1);
    const int x0i = (int)fminf(fmaxf(x0f,        0.0f), Wm1);
    const int x1i = (int)fminf(fmaxf(x0f + 1.0f, 0.0f), Wm1);

    // in-plane tap offsets, fixed for every channel
    const int oTL = y0i * FM_W + x0i;
    const int oTR = y0i * FM_W + x1i;
    const int oBL = y1i * FM_W + x0i;
    const int oBR = y1i * FM_W + x1i;

    const int c0 = cz * C_PER_BLK;
    const float* plane = fm + ((long long)b * FM_C + c0) * (long long)HW;
    float* op = out + ((long long)m * FM_C + c0) * (long long)NPOS + t;

    if (valid) {
        // warm the first taps' cachelines (gfx1250 global_prefetch_b8)
        __builtin_prefetch(plane + oTL, 0, 1);
        __builtin_prefetch(plane + oBL, 0, 1);

        #pragma unroll 4
        for (int c = 0; c < C_PER_BLK; ++c) {
            const float* p0 = plane + (long long)c * HW;
            if (c + PF_DIST < C_PER_BLK) {
                __builtin_prefetch(p0 + (long long)PF_DIST * HW + oTL, 0, 1);
                __builtin_prefetch(p0 + (long long)PF_DIST * HW + oBL, 0, 1);
            }
            const float tl = p0[oTL];
            const float tr = p0[oTR];
            const float bl = p0[oBL];
            const float br = p0[oBR];
            const float top = tl + (tr - tl) * xf;
            const float bot = bl + (br - bl) * xf;
            const float r   = top + (bot - top) * yf;
            __builtin_nontemporal_store(r, op);   // stream out, spare L2
            op += NPOS;
        }
    } else {
        // extrapolation value (0.0f) for every channel at this position
        #pragma unroll 4
        for (int c = 0; c < C_PER_BLK; ++c) {
            __builtin_nontemporal_store(0.0f, op);
            op += NPOS;
        }
    }
}

extern "C" void kernel_launch(void* const* d_in, const int* in_sizes, int n_in,
                              void* d_out, int out_size, void* d_ws, size_t ws_size,
                              hipStream_t stream) {
    (void)in_sizes; (void)n_in; (void)out_size; (void)d_ws; (void)ws_size;

    const float* fm      = (const float*)d_in[0];  // featuremap (4,256,200,336)
    const float* boxes   = (const float*)d_in[1];  // (1024,4)
    const int*   box_ind = (const int*)  d_in[2];  // (1024,)
    float*       out     = (float*)d_out;          // (1024,256,14,14)

    dim3 grid(M_BOXES, CH_SPLIT, 1);   // 8192 blocks
    dim3 block(224, 1, 1);             // 7 wave32s; threads 0..195 active
    crop_resize_gfx1250<<<grid, block, 0, stream>>>(fm, boxes, box_ind, out);
}